// BallQuery_86260123173793
// MI455X (gfx1250) — compile-verified
//
#include <hip/hip_runtime.h>

// ---------------------------------------------------------------------------
// BallQuery + grouping for MI455X (gfx1250, wave32).
//
// d2 = |c|^2 + |p|^2 - 2 c.p  via ONE v_wmma_f32_16x16x4_f32 per
// (16 centers x 16 points) tile:
//   A (16x4, const per wave) : row m = [-2cx, -2cy, -2cz, 1]
//   B (4x16, LDS, pipelined) : col n = [ px,  py,  pz, |p|^2 ]
//   C (seed, const per wave) : row m = |c_m|^2
//
// 8 waves / workgroup, 64 centers: wave pair (2g, 2g+1) owns centers
// [16g,16g+16); the even wave scans point indices [0,4096), the odd wave
// [4096,8192). Index-range split => the two ordered partial lists
// concatenate trivially (all low-half indices precede all high-half ones).
// Selection: ballot + popc-prefix, counts replicated in registers (the
// half-wave hit mask is uniform per half), wave-uniform skip of hitless
// rows, early exit per 32-tile group. Phase 2: L2-resident feature gather
// + non-temporal streaming stores (memory roofline ~12us @ 23.3 TB/s).
// ---------------------------------------------------------------------------

typedef float v2f __attribute__((ext_vector_type(2)));
typedef float v8f __attribute__((ext_vector_type(8)));

#if __has_builtin(__builtin_amdgcn_ballot_w32)
#define BALLOT32(p) __builtin_amdgcn_ballot_w32(p)
#else
#define BALLOT32(p) ((unsigned)__ballot(p))
#endif

#define RADIUS2 0.01f

constexpr int Bn = 8;      // batches
constexpr int Nn = 8192;   // points
constexpr int Mn = 2048;   // centers
constexpr int Cc = 64;     // feature channels
constexpr int Kk = 32;     // neighbors
constexpr int CPG = 64;    // centers per workgroup
constexpr int WAVES = 8;   // 2 scan-waves per 16-center group
constexpr int TPB = WAVES * 32;
constexpr int HTILES = (Nn / 16) / 2;  // 256 tiles per scan-wave
constexpr int TGRP = 32;               // tiles between early-exit checks

// dynamic LDS layout (points padded by 16 float4 so the B-operand prefetch
// one tile ahead never reads out of bounds; pad contents never consumed)
constexpr int PTS_PAD = Nn + 16;
constexpr size_t OFF_CENT = (size_t)PTS_PAD * 16;              // float4[64]
constexpr size_t OFF_CNT  = OFF_CENT + (size_t)CPG * 16;       // int[2][64]
constexpr size_t OFF_LIST = OFF_CNT + (size_t)2 * CPG * 4;     // int[2][64][32]
constexpr size_t SMEM_BYTES = OFF_LIST + (size_t)2 * CPG * Kk * 4;  // ~146KB

__global__ __launch_bounds__(TPB) void ball_query_group_kernel(
    const float* __restrict__ pts,   // (B,3,N)
    const float* __restrict__ cent,  // (B,3,M)
    const float* __restrict__ feat,  // (B,64,N)
    float* __restrict__ out)         // (B,67,M,32)
{
  extern __shared__ char smem[];
  float4* s_pts  = (float4*)smem;
  float4* s_cent = (float4*)(smem + OFF_CENT);
  int*    s_cnt  = (int*)(smem + OFF_CNT);   // [hsel][64]
  int*    s_list = (int*)(smem + OFF_LIST);  // [hsel][64][32]

  const int wg    = blockIdx.x;
  const int b     = wg / (Mn / CPG);
  const int mbase = (wg % (Mn / CPG)) * CPG;
  const int tid   = threadIdx.x;
  const int lane  = tid & 31;
  const int wave  = tid >> 5;
  const int hl    = lane >> 4;   // half-wave: K={0,1} vs K={2,3}, rows r vs r+8
  const int lidx  = lane & 15;
  const int grp   = wave >> 1;   // 16-center group (0..3)
  const int hsel  = wave & 1;    // 0: indices [0,4096), 1: [4096,8192)

  const float* pb = pts  + (size_t)b * 3 * Nn;
  const float* cb = cent + (size_t)b * 3 * Mn;
  const float* fb = feat + (size_t)b * Cc * Nn;

  // ---- stage points (+ |p|^2) and this tile's centers into LDS ----
  for (int n = tid; n < Nn; n += TPB) {
    float x = pb[n], y = pb[Nn + n], z = pb[2 * Nn + n];
    s_pts[n] = make_float4(x, y, z, x * x + y * y + z * z);
  }
  if (tid < CPG) {
    int m = mbase + tid;
    float x = cb[m], y = cb[Mn + m], z = cb[2 * Mn + m];
    s_cent[tid] = make_float4(x, y, z, x * x + y * y + z * z);
  }
  __syncthreads();

  // ---- constant WMMA operands for this wave's 16 centers ----
  // A layout (16x4 f32): lanes 0-15 hold K0/K1, lanes 16-31 hold K2/K3
  float4 ca = s_cent[grp * 16 + lidx];
  v2f a;
  a.x = hl ? (-2.0f * ca.z) : (-2.0f * ca.x);  // K=2 : K=0
  a.y = hl ? 1.0f           : (-2.0f * ca.y);  // K=3 : K=1
  // C seed: VGPR r holds row r (lanes 0-15) / row r+8 (lanes 16-31)
  v8f cinit;
#pragma unroll
  for (int r = 0; r < 8; ++r)
    cinit[r] = s_cent[grp * 16 + r + 8 * hl].w;

  const unsigned ltmask = (1u << lidx) - 1u;
  int* wlist = s_list + (hsel * CPG + grp * 16) * Kk;  // this wave's 16 lists
  int* wcnt  = s_cnt + hsel * CPG + grp * 16;

  // Per-half-wave replicated neighbor counts (hm uniform within a half).
  int cnt[8] = {0, 0, 0, 0, 0, 0, 0, 0};

  // ---- phase 1: scan this wave's half of the point indices, in order ----
  const int t0 = hsel * HTILES;  // first tile of this wave's range
  const char* bp =
      (const char*)s_pts + (size_t)(t0 * 16 + lidx) * 16 + (size_t)hl * 8;
  float2 pv = *(const float2*)bp;
  int n = t0 * 16 + lidx;  // this lane's current point index

  for (int g = 0; g < HTILES / TGRP; ++g) {
    for (int i = 0; i < TGRP; ++i) {
      float2 nx = *(const float2*)(bp + 256);  // prefetch next tile's B
      bp += 256;
      v2f bv;
      bv.x = pv.x;
      bv.y = pv.y;

      v8f d = __builtin_amdgcn_wmma_f32_16x16x4_f32(
          false, a, false, bv, (short)0, cinit, false, false);

#pragma unroll
      for (int r = 0; r < 8; ++r) {
        bool hit = d[r] < RADIUS2;
        unsigned bal = BALLOT32(hit);
        if (bal) {  // wave-uniform: skip hitless rows fast
          unsigned hm = hl ? (bal >> 16) : (bal & 0xffffu);
          int base = cnt[r];
          if (hit) {
            int pos = base + __popc(hm & ltmask);
            if (pos < Kk) wlist[(r + 8 * hl) * Kk + pos] = n;
          }
          cnt[r] = base + __popc(hm);
        }
      }
      n += 16;
      pv = nx;
    }
    // uniform early exit once all 16 centers of this wave are full
    bool full = (cnt[0] >= Kk) & (cnt[1] >= Kk) & (cnt[2] >= Kk) &
                (cnt[3] >= Kk) & (cnt[4] >= Kk) & (cnt[5] >= Kk) &
                (cnt[6] >= Kk) & (cnt[7] >= Kk);
    if (BALLOT32(full) == 0xffffffffu) break;
  }

  // flush counts (lane 0 -> rows 0..7, lane 16 -> rows 8..15)
  if (lidx == 0) {
#pragma unroll
    for (int r = 0; r < 8; ++r) wcnt[r + 8 * hl] = cnt[r];
  }
  __syncthreads();  // lists/counts are merged across wave pairs below

  // ---- phase 2: concat partial lists, pad, gather, write (B,67,M,32) ----
  // 8 waves x 8 centers each.
  const size_t obase = (size_t)b * 67 * Mn * Kk;
  for (int t = 0; t < CPG / WAVES; ++t) {
    int mloc = wave * (CPG / WAVES) + t;
    int m = mbase + mloc;
    const int* lA = s_list + mloc * Kk;          // low-index half
    const int* lB = s_list + (CPG + mloc) * Kk;  // high-index half
    int cA = s_cnt[mloc];
    cA = cA > Kk ? Kk : cA;            // if ==Kk, cA is exact (<=32 found low)
    int cBr = s_cnt[CPG + mloc];
    int need = Kk - cA;
    int cB = cBr < need ? cBr : need;
    int total = cA + cB;

    int idx0 = (cA > 0) ? lA[0] : ((cB > 0) ? lB[0] : 0);
    int idx;
    if (lane < cA)
      idx = lA[lane];
    else if (lane < total)
      idx = lB[lane - cA];
    else
      idx = idx0;

    float4 p = s_pts[idx];
    float4 c = s_cent[mloc];
    size_t o = obase + (size_t)m * Kk + lane;
    __builtin_nontemporal_store(p.x - c.x, out + o);
    __builtin_nontemporal_store(p.y - c.y, out + o + (size_t)Mn * Kk);
    __builtin_nontemporal_store(p.z - c.z, out + o + (size_t)2 * Mn * Kk);

    const float* f = fb + idx;
    float* oo = out + o + (size_t)3 * Mn * Kk;
#pragma unroll 4
    for (int ch = 0; ch < Cc; ++ch)
      __builtin_nontemporal_store(f[(size_t)ch * Nn],
                                  oo + (size_t)ch * Mn * Kk);
  }
}

extern "C" void kernel_launch(void* const* d_in, const int* in_sizes, int n_in,
                              void* d_out, int out_size, void* d_ws,
                              size_t ws_size, hipStream_t stream) {
  const float* pts  = (const float*)d_in[0];  // (8,3,8192)
  const float* cent = (const float*)d_in[1];  // (8,3,2048)
  const float* feat = (const float*)d_in[2];  // (8,64,8192)
  float* out = (float*)d_out;                 // (8,67,2048,32)

  dim3 grid(Bn * (Mn / CPG));  // 256 workgroups
  ball_query_group_kernel<<<grid, TPB, SMEM_BYTES, stream>>>(pts, cent, feat,
                                                             out);
}